// FastAttention_33294586479005
// MI455X (gfx1250) — compile-verified
//
#include <hip/hip_runtime.h>
#include <hip/hip_bf16.h>

typedef float v2f __attribute__((ext_vector_type(2)));
typedef float v8f __attribute__((ext_vector_type(8)));

namespace {
constexpr int B = 2, S = 1024, D = 512, H = 8, DK = 64, R = 16;
constexpr int G = 2, GD = 32, NH = 4, K_MAX = 64;
constexpr float BW_INV = 0.25f;      // 1 / lsh_bandwidth
constexpr float NEGV = -1e30f;
constexpr int SW = S / 32;           // 32-bit words per S-length bitmask
constexpr int KPAD = 33;             // padded row stride (floats) for k panels

// order-preserving float -> u32 key (monotone increasing)
__device__ __forceinline__ unsigned fkey(float f) {
  unsigned u = __float_as_uint(f);
  return (u & 0x80000000u) ? ~u : (u | 0x80000000u);
}
__device__ __forceinline__ bool any_byte_eq(unsigned a, unsigned b) {
  unsigned x = a ^ b;
  return ((x - 0x01010101u) & ~x & 0x80808080u) != 0u;
}
__device__ __forceinline__ unsigned ballot32(bool p) {
  return (unsigned)__ballot((int)p);
}
// DS-space byte offset of an LDS pointer (what async-to-LDS VDST takes)
__device__ __forceinline__ unsigned lds_addr(const void* p) {
  return (unsigned)(unsigned long long)
      (__attribute__((address_space(3))) const char*)p;
}
// 16B-per-lane async global->LDS copy (ASYNCcnt-tracked)
__device__ __forceinline__ void async_copy_b128(unsigned lds_off, const void* gp) {
  asm volatile("global_load_async_to_lds_b128 %0, %1, off"
               :: "v"(lds_off), "v"((unsigned long long)gp)
               : "memory");
}
} // namespace

// ---------------- zero the union bitmask ----------------
__global__ __launch_bounds__(256) void zero_union_kernel(unsigned* __restrict__ uni) {
  int i = blockIdx.x * 256 + threadIdx.x;
  if (i < B * H * S * SW) uni[i] = 0u;
}

// ---------------- q/k low-rank + v projections ----------------
// one block per (b,h,s); 64 threads (one per DK output)
__global__ __launch_bounds__(64) void proj_kernel(
    const float* __restrict__ q, const float* __restrict__ k, const float* __restrict__ v,
    const float* __restrict__ Aq, const float* __restrict__ Bq,
    const float* __restrict__ Ak, const float* __restrict__ Bk,
    const float* __restrict__ Wv,
    float* __restrict__ qup, float* __restrict__ kup, float* __restrict__ vup) {
  __shared__ float qL[D], kL[D], vL[D];
  __shared__ float tq[R], tk[R];
  int idx = blockIdx.x;
  int s = idx & (S - 1);
  int h = (idx >> 10) & (H - 1);
  int b = idx >> 13;
  int t = threadIdx.x;

  const float* qrow = q + ((size_t)b * S + s) * D;
  const float* krow = k + ((size_t)b * S + s) * D;
  const float* vrow = v + ((size_t)b * S + s) * D;
  for (int d = t; d < D; d += 64) { qL[d] = qrow[d]; kL[d] = krow[d]; vL[d] = vrow[d]; }
  __syncthreads();

  if (t < R) {
    float a = 0.f;
    for (int d = 0; d < D; ++d) a += qL[d] * Aq[((size_t)h * D + d) * R + t];
    tq[t] = a;
  } else if (t < 2 * R) {
    int r = t - R;
    float a = 0.f;
    for (int d = 0; d < D; ++d) a += kL[d] * Ak[((size_t)h * D + d) * R + r];
    tk[r] = a;
  }
  __syncthreads();

  float qo = 0.f, ko = 0.f, vo = 0.f;
  for (int r = 0; r < R; ++r) {
    qo += tq[r] * Bq[((size_t)h * R + r) * DK + t];
    ko += tk[r] * Bk[((size_t)h * R + r) * DK + t];
  }
  for (int d = 0; d < D; ++d) vo += vL[d] * Wv[((size_t)h * D + d) * DK + t];

  size_t o = (((size_t)b * H + h) * S + s) * DK + t;
  qup[o] = qo; kup[o] = ko; vup[o] = vo;
}

// ---------------- LSH bucket hashes (packed 4x8b) + sign masks ----------------
__global__ __launch_bounds__(256) void hash_kernel(
    const float* __restrict__ qup, const float* __restrict__ kup,
    const float* __restrict__ lsh,
    unsigned* __restrict__ hq, unsigned* __restrict__ hk,
    unsigned* __restrict__ smq, unsigned* __restrict__ smk) {
  int t = blockIdx.x * 256 + threadIdx.x;
  if (t >= B * H * G * S) return;
  int s = t & (S - 1);
  int g = (t >> 10) & (G - 1);
  int h = (t >> 11) & (H - 1);
  int b = t >> 14;

  const float* qg = qup + (((size_t)b * H + h) * S + s) * DK + g * GD;
  const float* kg = kup + (((size_t)b * H + h) * S + s) * DK + g * GD;
  const float* pr = lsh + (size_t)(h * G + g) * GD * NH;

  float aq[NH] = {0.f, 0.f, 0.f, 0.f}, ak[NH] = {0.f, 0.f, 0.f, 0.f};
  unsigned mq = 0u, mk = 0u;
  for (int d = 0; d < GD; ++d) {
    float qv = qg[d], kv = kg[d];
    if (qv > 0.f) mq |= 1u << d;
    if (kv > 0.f) mk |= 1u << d;
    for (int n = 0; n < NH; ++n) {
      float p = pr[d * NH + n];
      aq[n] += qv * p; ak[n] += kv * p;
    }
  }
  unsigned pq = 0u, pk = 0u;
  for (int n = 0; n < NH; ++n) {
    float f = floorf(aq[n] * BW_INV), m = fmodf(f, 64.f);
    if (m < 0.f) m += 64.f;
    pq |= ((unsigned)(int)m) << (8 * n);
    f = floorf(ak[n] * BW_INV); m = fmodf(f, 64.f);
    if (m < 0.f) m += 64.f;
    pk |= ((unsigned)(int)m) << (8 * n);
  }
  hq[t] = pq; hk[t] = pk;
  if (b == B - 1) {
    int o = (h * G + g) * S + s;
    smq[o] = mq; smk[o] = mk;
  }
}

// ---------------- WMMA group similarity + top-k selection ----------------
// one block (8 waves) per (b,h,g, 16-row strip)
// k panels double-buffered in LDS via global_load_async_to_lds_b128
__global__ __launch_bounds__(256) void simselect_kernel(
    const float* __restrict__ qup, const float* __restrict__ kup,
    const unsigned* __restrict__ hq, const unsigned* __restrict__ hk,
    const unsigned* __restrict__ smq, const unsigned* __restrict__ smk,
    unsigned* __restrict__ uni) {
  extern __shared__ unsigned smem[];
  unsigned* keys  = smem;               // [16][S]  order keys
  unsigned* cmask = keys + 16 * S;      // [16][SW] combined bitmask
  unsigned* hqs   = cmask + 16 * SW;    // [16]     packed row hashes
  unsigned* sqs   = hqs + 16;           // [16]     row sign masks (last batch)
  float* kst = (float*)(sqs + 16);      // [8 waves][2][16*KPAD] k panels

  int tile = blockIdx.x;
  int i0 = (tile & 63) * 16; tile >>= 6;
  int g = tile & (G - 1);    tile >>= 1;
  int h = tile & (H - 1);    tile >>= 3;
  int b = tile;

  int lane = threadIdx.x & 31, wave = threadIdx.x >> 5;
  for (int i = threadIdx.x; i < 16 * SW; i += 256) cmask[i] = 0u;
  if (threadIdx.x < 16) {
    int i = i0 + threadIdx.x;
    hqs[threadIdx.x] = hq[(((size_t)b * H + h) * G + g) * S + i];
    sqs[threadIdx.x] = smq[((size_t)h * G + g) * S + i];
  }
  __syncthreads();

  const int koff = 2 * (lane >> 4);
  const int mrow = lane & 15;
  const size_t bhS = ((size_t)b * H + h) * S;

  // A (16x32 strip of qg) stays resident across all N tiles
  const float* abase = qup + (bhS + i0 + mrow) * DK + g * GD;
  v2f a[8];
  for (int kk = 0; kk < 8; ++kk) {
    a[kk].x = abase[4 * kk + koff];
    a[kk].y = abase[4 * kk + koff + 1];
  }

  float* bufs = kst + wave * 2 * (16 * KPAD);
  // async-stage one 16x32 k panel: 4 instrs x (32 lanes x 16B), 4 rows each
  auto issue_panel = [&](int jt, float* buf) {
    int rb = lane >> 3;          // base row 0..3
    int ch = lane & 7;           // 16B chunk within 128B row
    int j0 = jt * 16;
    for (int i = 0; i < 4; ++i) {
      int row = i * 4 + rb;
      const float* gp = kup + (bhS + j0 + row) * DK + g * GD + ch * 4;
      async_copy_b128(lds_addr(buf + row * KPAD + ch * 4), gp);
    }
  };

  issue_panel(wave, bufs);
  int pp = 0;
  for (int jt = wave; jt < S / 16; jt += 8) {
    float* cur = bufs + pp * (16 * KPAD);
    if (jt + 8 < S / 16) {
      issue_panel(jt + 8, bufs + (pp ^ 1) * (16 * KPAD));
      asm volatile("s_wait_asynccnt 0x4" ::: "memory");  // current panel landed
    } else {
      asm volatile("s_wait_asynccnt 0x0" ::: "memory");
    }
    int j0 = jt * 16;
    v8f acc = {0.f, 0.f, 0.f, 0.f, 0.f, 0.f, 0.f, 0.f};
    const float* brow = cur + mrow * KPAD;
    for (int kk = 0; kk < 8; ++kk) {
      v2f bm;
      bm.x = brow[4 * kk + koff];
      bm.y = brow[4 * kk + koff + 1];
      acc = __builtin_amdgcn_wmma_f32_16x16x4_f32(false, a[kk], false, bm,
                                                  (short)0, acc, false, false);
    }
    unsigned hkl = hk[(((size_t)b * H + h) * G + g) * S + j0 + mrow];
    unsigned skl = smk[((size_t)h * G + g) * S + j0 + mrow];
    int half = (j0 >> 4) & 1, word = j0 >> 5;
    for (int c = 0; c < 8; ++c) {
      int m = c + 8 * (lane >> 4);  // strip row of this lane's C element
      bool comb = any_byte_eq(hqs[m], hkl) && (sqs[m] == skl);
      keys[m * S + j0 + mrow] = comb ? fkey(acc[c]) : fkey(NEGV);
      unsigned bb = ballot32(comb);
      if (lane == 0) {
        unsigned lo = bb & 0xFFFFu, hi = bb >> 16;
        if (lo) atomicOr(&cmask[c * SW + word], lo << (16 * half));
        if (hi) atomicOr(&cmask[(c + 8) * SW + word], hi << (16 * half));
      }
    }
    pp ^= 1;
  }
  __syncthreads();

  // per-row exact kth-largest via 32-step binary search on order keys
  for (int rr = 0; rr < 2; ++rr) {
    int r = wave + rr * 8;
    const unsigned* krow = keys + r * S;
    unsigned prefix = 0u;
    for (int bit = 31; bit >= 0; --bit) {
      unsigned cv = prefix | (1u << bit);
      int cnt = 0;
      for (int j = lane; j < S; j += 32) cnt += (krow[j] >= cv) ? 1 : 0;
      for (int off = 16; off; off >>= 1) cnt += __shfl_xor(cnt, off, 32);
      if (cnt >= K_MAX) prefix = cv;
    }
    unsigned* urow = uni + (bhS + i0 + r) * SW;
    for (int w = 0; w < SW; ++w) {
      int j = w * 32 + lane;
      bool sel = ((cmask[r * SW + w] >> lane) & 1u) && (krow[j] >= prefix);
      unsigned bits = ballot32(sel);
      if (lane == 0 && bits) atomicOr(&urow[w], bits);
    }
  }
}

// ---------------- union bitmask -> first K_MAX ascending candidates ----------------
__global__ __launch_bounds__(256) void cand_kernel(const unsigned* __restrict__ uni,
                                                   int* __restrict__ cand) {
  int row = blockIdx.x * 8 + (threadIdx.x >> 5);  // (b*H+h)*S + i
  int lane = threadIdx.x & 31;
  const unsigned* urow = uni + (size_t)row * SW;
  int* crow = cand + (size_t)row * K_MAX;
  int base = 0;
  for (int w = 0; w < SW; ++w) {
    unsigned word = urow[w];
    if ((word >> lane) & 1u) {
      int rank = base + __popc(word & ((1u << lane) - 1u));
      if (rank < K_MAX) crow[rank] = w * 32 + lane;
    }
    base += __popc(word);
  }
  for (int rk = base + lane; rk < K_MAX; rk += 32) crow[rk] = -1;
}

// ---------------- sparse softmax attention over candidates ----------------
__global__ __launch_bounds__(256) void attn_kernel(
    const float* __restrict__ qup, const float* __restrict__ kup,
    const float* __restrict__ vup, const int* __restrict__ cand,
    float* __restrict__ outh) {
  __shared__ float pS[8][K_MAX];
  __shared__ int   jS[8][K_MAX];
  __shared__ float qS[8][DK];
  int wave = threadIdx.x >> 5, lane = threadIdx.x & 31;
  int row = blockIdx.x * 8 + wave;   // (b*H+h)*S + i
  size_t bh = (size_t)(row >> 10);   // b*H+h

  const float* qrow = qup + (size_t)row * DK;
  qS[wave][lane] = qrow[lane];
  qS[wave][lane + 32] = qrow[lane + 32];
  int c0 = cand[(size_t)row * K_MAX + lane];
  int c1 = cand[(size_t)row * K_MAX + 32 + lane];
  int j0c = c0 < 0 ? 0 : c0;
  int j1c = c1 < 0 ? 0 : c1;
  jS[wave][lane] = j0c;
  jS[wave][lane + 32] = j1c;
  __syncthreads();

  const float* k0 = kup + (bh * S + j0c) * DK;
  const float* k1 = kup + (bh * S + j1c) * DK;
  float s0 = 0.f, s1 = 0.f;
  for (int d = 0; d < DK; ++d) {
    float qd = qS[wave][d];
    s0 += qd * k0[d]; s1 += qd * k1[d];
  }
  float l0 = (c0 >= 0) ? s0 * 0.125f : NEGV;
  float l1 = (c1 >= 0) ? s1 * 0.125f : NEGV;
  float mx = fmaxf(l0, l1);
  for (int off = 16; off; off >>= 1) mx = fmaxf(mx, __shfl_xor(mx, off, 32));
  float e0 = (c0 >= 0) ? expf(l0 - mx) : 0.f;
  float e1 = (c1 >= 0) ? expf(l1 - mx) : 0.f;
  float sum = e0 + e1;
  for (int off = 16; off; off >>= 1) sum += __shfl_xor(sum, off, 32);
  float inv = sum > 0.f ? 1.f / sum : 0.f;
  pS[wave][lane] = e0 * inv;
  pS[wave][lane + 32] = e1 * inv;
  __syncthreads();

  float o0 = 0.f, o1 = 0.f;
  for (int kk = 0; kk < K_MAX; ++kk) {
    float p = pS[wave][kk];
    const float* vr = vup + (bh * S + jS[wave][kk]) * DK;
    o0 += p * vr[lane];
    o1 += p * vr[lane + 32];
  }
  outh[(size_t)row * DK + lane] = o0;
  outh[(size_t)row * DK + lane + 32] = o1;
}

// ---------------- WMMA output projection: [B*S,H*DK] x Wo[H*DK,D] ----------------
// one block per (n-panel, 8 m-tiles); Wo panel (512x16) async-staged to LDS once
__global__ __launch_bounds__(256) void outproj_kernel(
    const float* __restrict__ outh, const float* __restrict__ Wo,
    float* __restrict__ out) {
  __shared__ float WoL[(H * DK) * 16];  // 32 KB: [k][n] panel
  int wave = threadIdx.x >> 5, lane = threadIdx.x & 31;
  int n0 = (blockIdx.x & 31) * 16;
  int m0 = ((blockIdx.x >> 5) * 8 + wave) * 16;

  // stage Wo[:, n0:n0+16]: 64 async instrs total, 8 per wave (8 rows each)
  {
    int rb = lane >> 2;       // base row 0..7
    int ch = lane & 3;        // 16B chunk within 64B row
    for (int i = 0; i < 8; ++i) {
      int k = (wave * 8 + i) * 8 + rb;
      const float* gp = Wo + (size_t)k * D + n0 + ch * 4;
      async_copy_b128(lds_addr(WoL + k * 16 + ch * 4), gp);
    }
    asm volatile("s_wait_asynccnt 0x0" ::: "memory");
  }
  __syncthreads();

  int row = m0 + (lane & 15);  // b*S + s
  int b = row >> 10, s = row & (S - 1);
  int koff = 2 * (lane >> 4);
  int nl = lane & 15;

  v8f acc = {0.f, 0.f, 0.f, 0.f, 0.f, 0.f, 0.f, 0.f};
  for (int kk = 0; kk < (H * DK) / 4; ++kk) {
    int k0 = 4 * kk + koff;
    v2f a, bm;
    a.x = outh[(((size_t)b * H + (k0 >> 6)) * S + s) * DK + (k0 & 63)];
    a.y = outh[(((size_t)b * H + ((k0 + 1) >> 6)) * S + s) * DK + ((k0 + 1) & 63)];
    bm.x = WoL[k0 * 16 + nl];
    bm.y = WoL[(k0 + 1) * 16 + nl];
    acc = __builtin_amdgcn_wmma_f32_16x16x4_f32(false, a, false, bm,
                                                (short)0, acc, false, false);
  }
  for (int c = 0; c < 8; ++c) {
    int m = m0 + c + 8 * (lane >> 4);
    out[(size_t)m * D + n0 + nl] = acc[c];
  }
}

// ---------------- host launcher ----------------
extern "C" void kernel_launch(void* const* d_in, const int* in_sizes, int n_in,
                              void* d_out, int out_size, void* d_ws, size_t ws_size,
                              hipStream_t stream) {
  const float* q   = (const float*)d_in[0];
  const float* k   = (const float*)d_in[1];
  const float* v   = (const float*)d_in[2];
  const float* Aq  = (const float*)d_in[3];
  const float* Bq  = (const float*)d_in[4];
  const float* Ak  = (const float*)d_in[5];
  const float* Bk  = (const float*)d_in[6];
  const float* Wv  = (const float*)d_in[7];
  const float* Wo  = (const float*)d_in[8];
  const float* lsh = (const float*)d_in[9];

  char* ws = (char*)d_ws;
  size_t off = 0;
  auto alloc = [&](size_t bytes) -> void* {
    void* p = ws + off;
    off += (bytes + 255) & ~(size_t)255;
    return p;
  };
  float*    qup  = (float*)alloc(sizeof(float) * (size_t)B * H * S * DK);
  float*    kup  = (float*)alloc(sizeof(float) * (size_t)B * H * S * DK);
  float*    vup  = (float*)alloc(sizeof(float) * (size_t)B * H * S * DK);
  float*    outh = (float*)alloc(sizeof(float) * (size_t)B * H * S * DK);
  unsigned* hq   = (unsigned*)alloc(sizeof(unsigned) * (size_t)B * H * G * S);
  unsigned* hk   = (unsigned*)alloc(sizeof(unsigned) * (size_t)B * H * G * S);
  unsigned* smq  = (unsigned*)alloc(sizeof(unsigned) * (size_t)H * G * S);
  unsigned* smk  = (unsigned*)alloc(sizeof(unsigned) * (size_t)H * G * S);
  unsigned* uni  = (unsigned*)alloc(sizeof(unsigned) * (size_t)B * H * S * SW);
  int*      cnd  = (int*)alloc(sizeof(int) * (size_t)B * H * S * K_MAX);

  zero_union_kernel<<<(B * H * S * SW + 255) / 256, 256, 0, stream>>>(uni);
  proj_kernel<<<B * H * S, 64, 0, stream>>>(q, k, v, Aq, Bq, Ak, Bk, Wv, qup, kup, vup);
  hash_kernel<<<(B * H * G * S + 255) / 256, 256, 0, stream>>>(qup, kup, lsh, hq, hk, smq, smk);
  size_t smem = (size_t)(16 * S + 16 * SW + 32) * sizeof(unsigned)
              + (size_t)8 * 2 * (16 * KPAD) * sizeof(float);
  simselect_kernel<<<B * H * G * (S / 16), 256, smem, stream>>>(qup, kup, hq, hk, smq, smk, uni);
  cand_kernel<<<B * H * S / 8, 256, 0, stream>>>(uni, cnd);
  attn_kernel<<<B * H * S / 8, 256, 0, stream>>>(qup, kup, vup, cnd, outh);
  outproj_kernel<<<((B * S / 16) * (D / 16)) / 8, 256, 0, stream>>>(outh, Wo, (float*)d_out);
}